// Energy4D_60464549593605
// MI455X (gfx1250) — compile-verified
//
#include <hip/hip_runtime.h>
#include <stdint.h>

// ---------------------------------------------------------------------------
// Energy4D hash-grid encoder for gfx1250 (MI455X).
// Gather-bound: per point, 4 encodings x 16 levels x 8 corner gathers (~134M
// random 8B gathers total). Fetches use the CDNA5 async-memory path with
// double-buffered software pipelining:
//   issue level l+1:  global_load_async_to_lds_b64 (per-lane gather, ASYNCcnt)
//   s_wait_asynccnt <= 8   (in-order completion => level l's 8 loads landed)
//   consume level l:  ds_load_b64 + trilinear FMA
// ---------------------------------------------------------------------------

#define HMASK 524287u        // HMAP - 1, HMAP = 1<<19
#define PRIME1 2654435761u
#define PRIME2 805459861u

__device__ __forceinline__ float clamp01(float v) {
    return fminf(fmaxf(v, 0.0f), 1.0f);
}

__device__ __forceinline__ int level_offset(int l) {
    // level sizes: l0 = 32768 (32^3), l1 = 262144 (64^3), l>=2 = 524288 (HMAP)
    return (l == 0) ? 0 : (l == 1) ? 32768 : (294912 + (l - 2) * 524288);
}

struct W3 { float wx, wy, wz; };

// Compute corner indices for one level and issue the 8 async gathers into the
// LDS buffer at byte offset `ldsbase` (this lane's slot; corner c at +c*256).
__device__ __forceinline__ void issue_level(const char* __restrict__ tab, int l,
                                            float u0, float u1, float u2,
                                            uint32_t ldsbase, W3* w) {
    const int res = 32 << l;
    const int off = level_offset(l);
    const float rm1 = (float)(res - 1);
    const float px = u0 * rm1;
    const float py = u1 * rm1;
    const float pz = u2 * rm1;
    // u in [0,1] so pos >= 0: trunc == floor
    const int p0x = min((int)px, res - 2);
    const int p0y = min((int)py, res - 2);
    const int p0z = min((int)pz, res - 2);
    w->wx = px - (float)p0x;
    w->wy = py - (float)p0y;
    w->wz = pz - (float)p0z;

    if (l >= 2) {
        // Strength-reduced spatial hash: 2 muls per level, 2 XORs per corner.
        const uint32_t hx0 = (uint32_t)p0x,          hx1 = hx0 + 1u;
        const uint32_t hy0 = (uint32_t)p0y * PRIME1, hy1 = hy0 + PRIME1;
        const uint32_t hz0 = (uint32_t)p0z * PRIME2, hz1 = hz0 + PRIME2;
#pragma unroll
        for (int c = 0; c < 8; ++c) {
            const uint32_t h = ((c & 1) ? hx1 : hx0)
                             ^ ((c & 2) ? hy1 : hy0)
                             ^ ((c & 4) ? hz1 : hz0);
            const int idx = (int)(h & HMASK) + off;
            const char* ga = tab + ((uint64_t)(uint32_t)idx << 3);
            const uint32_t la = ldsbase + (uint32_t)(c * 256);
            asm volatile("global_load_async_to_lds_b64 %0, %1, off"
                         :
                         : "v"(la), "v"(ga)
                         : "memory");
        }
    } else {
        const int base = p0x + res * (p0y + res * p0z) + off;
        const int rr = res * res;
#pragma unroll
        for (int c = 0; c < 8; ++c) {
            const int idx = base + (c & 1) + ((c & 2) ? res : 0) + ((c & 4) ? rr : 0);
            const char* ga = tab + ((uint64_t)(uint32_t)idx << 3);
            const uint32_t la = ldsbase + (uint32_t)(c * 256);
            asm volatile("global_load_async_to_lds_b64 %0, %1, off"
                         :
                         : "v"(la), "v"(ga)
                         : "memory");
        }
    }
}

// Trilinear-weighted accumulation of the 8 staged corners.
__device__ __forceinline__ float2 consume_level(const float2* __restrict__ slot,
                                                const W3& w) {
    const float ix = 1.0f - w.wx;
    const float iy = 1.0f - w.wy;
    const float iz = 1.0f - w.wz;
    float a0 = 0.0f, a1 = 0.0f;
#pragma unroll
    for (int c = 0; c < 8; ++c) {
        const float2 f = slot[c * 32];
        const float wt = ((c & 1) ? w.wx : ix)
                       * ((c & 2) ? w.wy : iy)
                       * ((c & 4) ? w.wz : iz);
        a0 = fmaf(f.x, wt, a0);
        a1 = fmaf(f.y, wt, a1);
    }
    return make_float2(a0, a1);
}

// Encode one 3D coordinate against one table; writes 16 float2 (one per level).
// Double-buffered pipeline: level l+1's gathers fly while level l is consumed.
__device__ __forceinline__ void encode3(const char* __restrict__ tab,
                                        float u0, float u1, float u2,
                                        float2* __restrict__ outp,
                                        uint32_t lds0,
                                        const float2* __restrict__ slot0) {
    W3 wcur, wnxt;
    issue_level(tab, 0, u0, u1, u2, lds0, &wcur);
#pragma unroll 1
    for (int l = 0; l < 16; ++l) {
        const int b = l & 1;
        if (l < 15) {
            issue_level(tab, l + 1, u0, u1, u2,
                        lds0 + (uint32_t)((b ^ 1) * 2048), &wnxt);
            // 16 async loads outstanding; first 8 (level l) complete in order.
            asm volatile("s_wait_asynccnt 0x8" ::: "memory");
        } else {
            asm volatile("s_wait_asynccnt 0x0" ::: "memory");
        }
        outp[l] = consume_level(slot0 + b * 256, wcur);
        wcur = wnxt;
    }
}

__global__ void __launch_bounds__(256)
energy4d_kernel(const float* __restrict__ coords,
                const float* __restrict__ refc,
                const char* __restrict__ xyz_tab,
                const char* __restrict__ xyt_tab,
                const char* __restrict__ yzt_tab,
                const char* __restrict__ xzt_tab,
                float* __restrict__ out,
                int n_pts) {
    // Per-wave staging: 2 buffers * 8 corners * 32 lanes * float2 = 4 KB.
    // 8 waves/block -> 32 KB of the 320 KB WGP LDS.
    __shared__ float2 smem[8 * 512];

    const int n = blockIdx.x * blockDim.x + threadIdx.x;
    if (n >= n_pts) return;

    const int wv = threadIdx.x >> 5;
    const int ln = threadIdx.x & 31;
    const float2* slot0 = &smem[wv * 512 + ln];
    // Generic pointer to LDS: low 32 bits are the workgroup-relative LDS byte
    // offset, which is what global_load_async_to_lds consumes.
    const uint32_t lds0 = (uint32_t)(uintptr_t)slot0;

    const float4 c4 = ((const float4*)coords)[n];
    const float4 r4 = ((const float4*)refc)[n];
    const float relx = c4.x - r4.x;
    const float rely = c4.y - r4.y;
    const float relz = c4.z - r4.z;
    const float relt = c4.w - r4.w;

    // _norm against LAT/LON/ELEV/TIME ranges (constant denominators ->
    // reciprocal multiplies; avoids v_div_scale/v_rcp IEEE-division chains).
    const float x = clamp01((relx + 90.0f)    * (1.0f / 180.0f));
    const float y = clamp01((rely + 180.0f)   * (1.0f / 360.0f));
    const float z = clamp01((relz + 11000.0f) * (1.0f / 20000.0f));
    const float t = clamp01(relt);
    const float ts = (t * 2.0f - 1.0f) * 0.9f;

    // hash_encode's u = clip((v + 1) / 2, 0, 1)
    const float ux = clamp01((x + 1.0f) * 0.5f);
    const float uy = clamp01((y + 1.0f) * 0.5f);
    const float uz = clamp01((z + 1.0f) * 0.5f);
    const float ut = clamp01((ts + 1.0f) * 0.5f);

    // Output layout per point: [spatial(16xf2) | xyt | yzt | xzt]
    float2* op = (float2*)out + (size_t)n * 64;
    encode3(xyz_tab, ux, uy, uz, op,      lds0, slot0);
    encode3(xyt_tab, ux, uy, ut, op + 16, lds0, slot0);
    encode3(yzt_tab, uy, uz, ut, op + 32, lds0, slot0);
    encode3(xzt_tab, ux, uz, ut, op + 48, lds0, slot0);
}

extern "C" void kernel_launch(void* const* d_in, const int* in_sizes, int n_in,
                              void* d_out, int out_size, void* d_ws, size_t ws_size,
                              hipStream_t stream) {
    const float* coords = (const float*)d_in[0];
    const float* refc   = (const float*)d_in[1];
    const char*  xyz    = (const char*)d_in[2];
    const char*  xyt    = (const char*)d_in[3];
    const char*  yzt    = (const char*)d_in[4];
    const char*  xzt    = (const char*)d_in[5];
    float* out = (float*)d_out;

    const int n_pts = in_sizes[0] / 4;  // coords is (N, 4) float32
    const int block = 256;
    const int grid = (n_pts + block - 1) / block;
    energy4d_kernel<<<grid, block, 0, stream>>>(coords, refc, xyz, xyt, yzt, xzt,
                                                out, n_pts);
}